// S4_naive_5823975653752
// MI455X (gfx1250) — compile-verified
//
#include <hip/hip_runtime.h>
#include <math.h>

// S4 layer for MI455X (gfx1250), fp32 WMMA path + CDNA5 async-to-LDS staging.
//
// Pipeline (all on `stream`, graph-capture safe):
//   1. build_aug        : M = [I - d/2*A | I + d/2*A | d*Bm]   (512 x 1025)
//   2. gj_step x512     : Gauss-Jordan elimination (no normalization)
//   3. finalize_solve   : Ad = diag^-1 * A2', Bd = diag^-1 * b'
//   4. krylov_v         : V[:,i] = Ad^i Bd, i=0..127   (127 seq matvecs)
//   5. gemm_wmma x7     : W = Ad^128 by repeated squaring (fp32 WMMA)
//   6. krylov_c         : Cmat[j,:] = C * W^j, j=0..63  (63 seq vecmats)
//   7. gemm_wmma        : kcoef[j*128+i] = Cmat(64x512) @ V(512x128)
//   8. conv_wmma        : y = X (*) k + D*X  as Toeplitz WMMA GEMM,
//                         k staged in LDS via global_load_async_to_lds_b128.

#define SN 512
#define SL 8192
#define SB 64
#define NC (2 * SN + 1)   // augmented width: A1 | A2 | rhs

typedef float v2f __attribute__((ext_vector_type(2)));
typedef float v8f __attribute__((ext_vector_type(8)));

// ---------------------------------------------------------------- setup
__global__ void build_aug(const float* __restrict__ A,
                          const float* __restrict__ Bm,
                          const float* __restrict__ log_delta,
                          float* __restrict__ M)
{
    // delta = exp(clamp(log_delta, log(1e-3), log(0.1)))
    float ld = log_delta[0];
    ld = fminf(fmaxf(ld, -6.9077552790f), -2.3025850930f);
    const float delta = expf(ld);

    const int i = blockIdx.x;                      // row
    for (int j = threadIdx.x; j < SN; j += blockDim.x) {
        const float a    = A[i * SN + j];
        const float diag = (i == j) ? 1.0f : 0.0f;
        M[i * NC + j]      = diag - 0.5f * delta * a;   // A1
        M[i * NC + SN + j] = diag + 0.5f * delta * a;   // A2
    }
    if (threadIdx.x == 0)
        M[i * NC + 2 * SN] = delta * Bm[i];             // rhs
}

// One elimination step: zero column k in every row i != k.
// Row k is read-only this step, so blocks are independent.
__global__ void gj_step(float* __restrict__ M, int k)
{
    const int i = blockIdx.x;
    if (i == k) return;                                 // uniform whole-block exit
    __shared__ float s_scale;
    if (threadIdx.x == 0)
        s_scale = M[i * NC + k] / M[k * NC + k];
    __syncthreads();
    const float sc = s_scale;
    // Columns < k of the A1 block are already zero in both rows.
    for (int j = k + threadIdx.x; j < NC; j += blockDim.x)
        M[i * NC + j] = fmaf(-sc, M[k * NC + j], M[i * NC + j]);
}

__global__ void finalize_solve(const float* __restrict__ M,
                               float* __restrict__ Ad,
                               float* __restrict__ Bd)
{
    const int i = blockIdx.x;
    const float inv = 1.0f / M[i * NC + i];
    for (int j = threadIdx.x; j < SN; j += blockDim.x)
        Ad[i * SN + j] = M[i * NC + SN + j] * inv;
    if (threadIdx.x == 0)
        Bd[i] = M[i * NC + 2 * SN] * inv;
}

// ---------------------------------------------------------------- Krylov
// V[n][i] = (Ad^i Bd)[n], i = 0..127. One workgroup, 512 threads.
__global__ void krylov_v(const float* __restrict__ Ad,
                         const float* __restrict__ Bd,
                         float* __restrict__ V)
{
    __shared__ float sv[SN];
    const int n = threadIdx.x;
    float v = Bd[n];
    sv[n] = v;
    V[n * 128 + 0] = v;
    for (int i = 1; i < 128; ++i) {
        __syncthreads();
        float acc = 0.0f;
        const float* __restrict__ row = Ad + n * SN;
        for (int m = 0; m < SN; ++m)
            acc = fmaf(row[m], sv[m], acc);
        __syncthreads();
        sv[n] = acc;
        V[n * 128 + i] = acc;
    }
}

// Cmat[j][n] = (C * W^j)[n], j = 0..63. One workgroup, 512 threads.
__global__ void krylov_c(const float* __restrict__ Cv,
                         const float* __restrict__ W,
                         float* __restrict__ Cmat)
{
    __shared__ float sc[SN];
    const int n = threadIdx.x;
    float c = Cv[n];
    sc[n] = c;
    Cmat[n] = c;
    for (int j = 1; j < 64; ++j) {
        __syncthreads();
        float acc = 0.0f;
        for (int m = 0; m < SN; ++m)
            acc = fmaf(sc[m], W[m * SN + n], acc);
        __syncthreads();
        sc[n] = acc;
        Cmat[j * SN + n] = acc;
    }
}

// ---------------------------------------------------------------- fp32 WMMA GEMM
// C(MxNn) = A(MxK) @ B(KxNn), row-major, one wave per 16x16 tile.
// Fragment layouts per CDNA5 ISA (16x16x4 f32):
//   A: lane<16 -> K={0,1}, lane>=16 -> K={2,3}; M = lane&15
//   B: mirrors C/D striping: VGPR i holds row K = 2*(lane>>4)+i, N = lane&15
//   C/D: VGPR r holds M = 8*(lane>>4)+r, N = lane&15
__global__ void gemm_wmma(const float* __restrict__ A,
                          const float* __restrict__ B,
                          float* __restrict__ Cm,
                          int M, int Nn, int K)
{
    const int lane  = threadIdx.x & 31;
    const int tn    = blockIdx.x * 16;
    const int tm    = blockIdx.y * 16;
    const int mrow  = lane & 15;
    const int khalf = (lane >> 4) << 1;

    v8f acc = {};
    for (int kk = 0; kk < K; kk += 4) {
        v2f a, b;
        const float* __restrict__ ap = A + (tm + mrow) * K + kk + khalf;
        a.x = ap[0];
        a.y = ap[1];
        const float* __restrict__ bp = B + (kk + khalf) * Nn + tn + mrow;
        b.x = bp[0];
        b.y = bp[Nn];
        acc = __builtin_amdgcn_wmma_f32_16x16x4_f32(
            false, a, false, b, (short)0, acc, false, false);
    }

    const int chalf = (lane >> 4) << 3;
    for (int r = 0; r < 8; ++r)
        Cm[(tm + chalf + r) * Nn + tn + mrow] = acc[r];
}

// ---------------------------------------------------------------- convolution
// y[b,t] = sum_{s<=t} X[b,s]*k[t-s] + D*X[b,t], as WMMA GEMM vs an
// on-the-fly lower-triangular Toeplitz operand built from k.
// k (32 KB) is staged into LDS with CDNA5 async DMA loads (ASYNCcnt path):
//   global_load_async_to_lds_b128 + s_wait_asynccnt 0 + workgroup barrier.
// Block: 256 threads = 8 waves; wave w owns t-tile 8*blockIdx.x+w,
// blockIdx.y owns a 16-row batch tile. Uniform control flow around WMMA.
__global__ void conv_wmma(const float* __restrict__ X,
                          const float* __restrict__ kcoef,
                          const float* __restrict__ D,
                          float* __restrict__ Y)
{
    __shared__ float sk[SL];                            // 32 KB of k

    // Async global->LDS DMA: 2048 x b128 transactions, 8 per thread.
    {
        const unsigned lds_base = (unsigned)(size_t)(&sk[0]);
        for (int idx = threadIdx.x; idx < SL / 4; idx += blockDim.x) {
            const unsigned lds_addr = lds_base + (unsigned)idx * 16u;
            const unsigned goff     = (unsigned)idx * 16u;
            asm volatile("global_load_async_to_lds_b128 %0, %1, %2"
                         :
                         : "v"(lds_addr), "v"(goff), "s"(kcoef)
                         : "memory");
        }
        asm volatile("s_wait_asynccnt 0" ::: "memory");
    }
    __syncthreads();

    const int lane  = threadIdx.x & 31;
    const int wave  = threadIdx.x >> 5;
    const int t0    = (blockIdx.x * 8 + wave) * 16;     // 0..8176
    const int b0    = blockIdx.y * 16;
    const int mrow  = lane & 15;
    const int khalf = (lane >> 4) << 1;

    v8f acc = {};
    const int send = t0 + 16;                           // s in [0, t0+15]
    for (int s0 = 0; s0 < send; s0 += 4) {
        const int s = s0 + khalf;                       // this lane's K rows
        v2f a, b;
        const float* __restrict__ xp = X + (b0 + mrow) * SL + s;
        // Speculative prefetch ~2 cachelines ahead along the X row; past-end
        // prefetches are silently dropped (ISA cache-control 7.3).
        __builtin_prefetch(xp + 128, 0, 1);
        a.x = xp[0];
        a.y = xp[1];
        const int d0 = t0 + mrow - s;                   // t - s for K-row 0
        b.x = (d0     >= 0) ? sk[d0]     : 0.0f;        // causal mask (select)
        b.y = (d0 - 1 >= 0) ? sk[d0 - 1] : 0.0f;
        acc = __builtin_amdgcn_wmma_f32_16x16x4_f32(
            false, a, false, b, (short)0, acc, false, false);
    }

    const float dv    = D[0];
    const int   chalf = (lane >> 4) << 3;
    for (int r = 0; r < 8; ++r) {
        const int bb = b0 + chalf + r;
        const int tt = t0 + mrow;
        Y[bb * SL + tt] = acc[r] + dv * X[bb * SL + tt];
    }
}

// ---------------------------------------------------------------- launcher
extern "C" void kernel_launch(void* const* d_in, const int* in_sizes, int n_in,
                              void* d_out, int out_size, void* d_ws, size_t ws_size,
                              hipStream_t stream)
{
    const float* X   = (const float*)d_in[0];   // [64, 8192]
    const float* A   = (const float*)d_in[1];   // [512, 512]
    const float* Bm  = (const float*)d_in[2];   // [512, 1]
    const float* Cv  = (const float*)d_in[3];   // [1, 512]
    const float* D   = (const float*)d_in[4];   // [1]
    const float* ldl = (const float*)d_in[5];   // [1]
    float* Y = (float*)d_out;

    float* ws   = (float*)d_ws;
    float* Maug = ws;                           // 512*1025
    float* Ad   = Maug + SN * NC;               // 512*512
    float* Bd   = Ad + SN * SN;                 // 512
    float* W0   = Bd + SN;                      // 512*512
    float* W1   = W0 + SN * SN;                 // 512*512
    float* V    = W1 + SN * SN;                 // 512*128
    float* Cmat = V + SN * 128;                 // 64*512
    float* kco  = Cmat + 64 * SN;               // 8192

    // 1-3: bilinear discretization via Gauss-Jordan
    build_aug<<<SN, 256, 0, stream>>>(A, Bm, ldl, Maug);
    for (int k = 0; k < SN; ++k)
        gj_step<<<SN, 256, 0, stream>>>(Maug, k);
    finalize_solve<<<SN, 256, 0, stream>>>(Maug, Ad, Bd);

    // 4: Krylov columns V[:,i] = Ad^i Bd
    krylov_v<<<1, SN, 0, stream>>>(Ad, Bd, V);

    // 5: W = Ad^128 via 7 squarings (fp32 WMMA, one wave per 16x16 tile)
    {
        float* cur = Ad;
        for (int it = 0; it < 7; ++it) {
            float* out = (it & 1) ? W1 : W0;
            gemm_wmma<<<dim3(SN / 16, SN / 16), 32, 0, stream>>>(
                cur, cur, out, SN, SN, SN);
            cur = out;
        }
        // cur == W0 after 7 iterations
        // 6: Cmat rows via W
        krylov_c<<<1, SN, 0, stream>>>(Cv, cur, Cmat);
    }

    // 7: kcoef[j*128+i] = (Cmat @ V)[j][i] = C Ad^(128j+i) Bd
    gemm_wmma<<<dim3(128 / 16, 64 / 16), 32, 0, stream>>>(
        Cmat, V, kco, 64, 128, SN);

    // 8: causal Toeplitz convolution + D*X passthrough
    conv_wmma<<<dim3(SL / 16 / 8, SB / 16), 256, 0, stream>>>(X, kco, D, Y);
}